// Channel_60876866454206
// MI455X (gfx1250) — compile-verified
//
#include <hip/hip_runtime.h>
#include <cstdint>
#include <cstddef>

// ---------------- static problem config (matches reference) ----------------
#define L_TAPS   8
#define N_BATCH  512
#define S_SYM    64
#define M_LEN    82
#define T_LEN    (S_SYM * M_LEN)          // 5248
#define NL_ROWS  (N_BATCH * L_TAPS)       // 4096
#define TWO_PI_F 6.283185307179586f
#define INV_SCS  (1.0f / 15000.0f)

typedef __attribute__((ext_vector_type(2))) float v2f;
typedef __attribute__((ext_vector_type(8))) float v8f;

typedef __attribute__((address_space(1))) int  gint_t;
typedef __attribute__((address_space(3))) int  lint_t;

// ---- optional CDNA5 async global->LDS path (guarded; fallback = plain load)
#if defined(__gfx1250__) && __has_builtin(__builtin_amdgcn_global_load_async_to_lds_b32)
#define HAVE_ASYNC_LDS 1
#else
#define HAVE_ASYNC_LDS 0
#endif

__device__ __forceinline__ void stage_f32(float* lds_dst, const float* g_src) {
#if HAVE_ASYNC_LDS
  __builtin_amdgcn_global_load_async_to_lds_b32(
      (gint_t*)g_src,     // global source (addrspace 1)
      (lint_t*)lds_dst,   // LDS destination (addrspace 3)
      0, 0);
#else
  *lds_dst = *g_src;
#endif
}

__device__ __forceinline__ void stage_wait() {
#if HAVE_ASYNC_LDS
#if __has_builtin(__builtin_amdgcn_s_wait_asynccnt)
  __builtin_amdgcn_s_wait_asynccnt(0);
#else
  asm volatile("s_wait_asynccnt 0" ::: "memory");
#endif
#endif
}

// ============================================================================
// Kernel A: per-(n,l) tap gains + Doppler phase ramp via WMMA outer product.
// One wave (32 threads) computes a 16x16 tile of phases[nl, s] = ph0[nl]*t[s]
// with V_WMMA_F32_16X16X4_F32 (A: ph0 in K-slot 0; B: t[s] in row 0), then
// per-lane sincos produces cof_fin stored to ws as [nl][s][re,im].
// ============================================================================
__global__ __launch_bounds__(32)
void coef_wmma_kernel(const float* __restrict__ ray_u,    // [N][L]
                      const float* __restrict__ angles_u, // [L]
                      const float* __restrict__ phase_u,  // [N][L]
                      const int*   __restrict__ velocity, // [1]
                      float* __restrict__ cf_ws)          // [NL_ROWS][S_SYM][2]
{
  const int lane = threadIdx.x;        // 0..31
  const int nl0  = blockIdx.x * 16;    // row tile base (multiple of 16 -> 2 n's)
  const int s0   = blockIdx.y * 16;    // col tile base

  // exponential power profile, normalized
  float pw[L_TAPS];
  float psum = 0.0f;
#pragma unroll
  for (int j = 0; j < L_TAPS; ++j) { pw[j] = __expf(-(float)j * 0.25f); psum += pw[j]; }
  const float inv_psum = 1.0f / psum;

  // every lane computes one of the 16 rows (lanes 16..31 duplicate rows 0..15
  // so addresses stay valid and EXEC stays all-ones for the WMMA)
  const int rlane = lane & 15;
  const int row   = nl0 + rlane;       // nl index
  const int n     = row >> 3;
  const int rank  = row & 7;           // descending-sort rank for cof,
                                       // original tap index for angle/phase

  // Rayleigh magnitudes for all 8 taps of this n, then stable rank-select
  float c[L_TAPS];
#pragma unroll
  for (int j = 0; j < L_TAPS; ++j) {
    float u = ray_u[n * L_TAPS + j];
    c[j] = sqrtf(pw[j] * inv_psum * 0.5f) * sqrtf(-2.0f * __logf(u));
  }
  float cof = 0.0f;
#pragma unroll
  for (int i = 0; i < L_TAPS; ++i) {
    int greater = 0;
#pragma unroll
    for (int j = 0; j < L_TAPS; ++j)
      greater += (c[j] > c[i]) || ((c[j] == c[i]) && (j < i));
    cof = (greater == rank) ? c[i] : cof;   // stable descending order
  }

  const float fd  = (float)(*velocity) * (3.6e9f / 3.0e8f);   // v/c * carrier
  const float ph0 = TWO_PI_F * __cosf(angles_u[rank] * TWO_PI_F) * fd
                  + phase_u[n * L_TAPS + rank] * TWO_PI_F;

  __shared__ float scof[16];
  if (lane < 16) scof[rlane] = cof;
  __syncthreads();

  // WMMA rank-1: A(16x4) has ph0 only in K=0 (vgpr0, lanes 0-15);
  // B(4x16) has t[s] only in row K=0 (vgpr0, lanes 0-15); rest zero.
  v2f a; a.x = (lane < 16) ? ph0 : 0.0f;                         a.y = 0.0f;
  v2f b; b.x = (lane < 16) ? ((float)(s0 + rlane) * INV_SCS) : 0.0f; b.y = 0.0f;
  v8f cacc = {0.0f, 0.0f, 0.0f, 0.0f, 0.0f, 0.0f, 0.0f, 0.0f};
  v8f d = __builtin_amdgcn_wmma_f32_16x16x4_f32(
      /*neg_a=*/false, a, /*neg_b=*/false, b,
      /*c_mod=*/(short)0, cacc, /*reuse_a=*/false, /*reuse_b=*/false);

  // D layout: vgpr v holds row v (lanes 0-15) / row v+8 (lanes 16-31), col = lane%16
  const int hi  = (lane >= 16) ? 8 : 0;
  const int col = s0 + rlane;
#pragma unroll
  for (int v = 0; v < 8; ++v) {
    const int   rl = v + hi;
    const float ph = d[v];
    float s_, c_;
    __sincosf(ph, &s_, &c_);
    const float cc  = scof[rl];
    const int   idx = ((nl0 + rl) * S_SYM + col) * 2;
    cf_ws[idx]     = cc * c_;   // Re{cof * e^{i ph}}
    cf_ws[idx + 1] = cc * s_;   // Im
  }
}

// ============================================================================
// Kernel B: H_t[n,s] = sum_l cof_fin[n,l,s], interleaved complex out.
// ============================================================================
__global__ __launch_bounds__(256)
void ht_kernel(const float* __restrict__ cf_ws, float* __restrict__ out_h) {
  const int i = blockIdx.x * blockDim.x + threadIdx.x;   // n*S + s
  if (i >= N_BATCH * S_SYM) return;
  const int n = i / S_SYM;
  const int s = i - n * S_SYM;
  float hr = 0.0f, hi = 0.0f;
#pragma unroll
  for (int l = 0; l < L_TAPS; ++l) {
    const int idx = ((n * L_TAPS + l) * S_SYM + s) * 2;
    hr += cf_ws[idx];
    hi += cf_ws[idx + 1];
  }
  out_h[i * 2]     = hr;
  out_h[i * 2 + 1] = hi;
}

// ============================================================================
// Kernel C: the HBM-bound 8-tap complex FIR with per-symbol taps.
// out[n,t] = (1/8) * sum_l x[n,t-l] * cf[n, l, (t-l)/82]   (zero for t<l)
// Block = (n, 1312-wide chunk); x(+7 halo) and the full 8x64 complex tap
// table are staged in LDS (async-to-LDS when available).
// ============================================================================
#define CHUNK 1312          // T_LEN / 4

__global__ __launch_bounds__(256)
void fir_kernel(const float* __restrict__ xre,
                const float* __restrict__ xim,
                const float* __restrict__ cf_ws,
                float* __restrict__ out) {
  __shared__ float xsr[CHUNK + 8];
  __shared__ float xsi[CHUNK + 8];
  __shared__ float cfs[L_TAPS * S_SYM * 2];   // 1024 floats

  const int n   = blockIdx.x;
  const int t0  = blockIdx.y * CHUNK;
  const int tid = threadIdx.x;

  const float* xrn = xre + (size_t)n * T_LEN;
  const float* xin = xim + (size_t)n * T_LEN;

  // stage x with 7-sample left halo (zero-fill before t=0)
  for (int i = tid; i < CHUNK + 7; i += 256) {
    const int t = t0 - 7 + i;
    if (t >= 0) {
      stage_f32(&xsr[i], &xrn[t]);
      stage_f32(&xsi[i], &xin[t]);
    } else {
      xsr[i] = 0.0f;
      xsi[i] = 0.0f;
    }
  }
  // stage this n's full tap table (contiguous 1024 floats)
  const float* cfn = cf_ws + (size_t)n * (L_TAPS * S_SYM * 2);
  for (int i = tid; i < L_TAPS * S_SYM * 2; i += 256)
    stage_f32(&cfs[i], &cfn[i]);

  stage_wait();
  __syncthreads();

  for (int j = tid; j < CHUNK; j += 256) {
    const int t    = t0 + j;
    const int s_hi = (int)((unsigned)t / (unsigned)M_LEN);
    const int m    = t - s_hi * M_LEN;        // t mod 82
    float or_ = 0.0f, oi_ = 0.0f;
#pragma unroll
    for (int l = 0; l < L_TAPS; ++l) {
      int sidx = (l <= m) ? s_hi : (s_hi - 1);
      sidx = (sidx < 0) ? 0 : sidx;           // x halo is zero there anyway
      const float xr = xsr[j + 7 - l];
      const float xi = xsi[j + 7 - l];
      const float cr = cfs[(l * S_SYM + sidx) * 2];
      const float ci = cfs[(l * S_SYM + sidx) * 2 + 1];
      or_ = fmaf(xr, cr, fmaf(-xi, ci, or_));
      oi_ = fmaf(xr, ci, fmaf(xi, cr, oi_));
    }
    const int o = (n * T_LEN + t) * 2;
    out[o]     = or_ * 0.125f;
    out[o + 1] = oi_ * 0.125f;
  }
}

// ============================================================================
extern "C" void kernel_launch(void* const* d_in, const int* in_sizes, int n_in,
                              void* d_out, int out_size, void* d_ws, size_t ws_size,
                              hipStream_t stream) {
  (void)in_sizes; (void)n_in; (void)out_size; (void)ws_size;

  const float* in_re  = (const float*)d_in[0];  // (N,1,T)
  const float* in_im  = (const float*)d_in[1];  // (N,1,T)
  const float* ray_u  = (const float*)d_in[2];  // (N,1,L)
  const float* angles = (const float*)d_in[3];  // (L,)
  const float* phase  = (const float*)d_in[4];  // (N,1,L,1)
  const int*   vel    = (const int*)d_in[5];    // scalar
  // d_in[6] = Ns (compile-time constant here)

  float* out   = (float*)d_out;                               // complex interleaved
  float* out_h = out + (size_t)N_BATCH * T_LEN * 2;           // H_t region
  float* cf_ws = (float*)d_ws;                                // 2 MB tap plane

  // A: coefficients + WMMA phase outer product
  dim3 gA(NL_ROWS / 16, S_SYM / 16);       // 256 x 4
  coef_wmma_kernel<<<gA, 32, 0, stream>>>(ray_u, angles, phase, vel, cf_ws);

  // B: H_t reduction
  ht_kernel<<<(N_BATCH * S_SYM + 255) / 256, 256, 0, stream>>>(cf_ws, out_h);

  // C: main FIR
  dim3 gC(N_BATCH, T_LEN / CHUNK);         // 512 x 4
  fir_kernel<<<gC, 256, 0, stream>>>(in_re, in_im, cf_ws, out);
}